// SparseLoRAMoE_44822278701274
// MI455X (gfx1250) — compile-verified
//
#include <hip/hip_runtime.h>
#include <hip/hip_bf16.h>

typedef __attribute__((ext_vector_type(2))) float v2f;
typedef __attribute__((ext_vector_type(8))) float v8f;

#define DIM   4096
#define NE    8
#define NR    16
#define NO    4096
#define NTOK  16384
#define CW    144     // 128 h cols + 8 logit cols + 8 pad
#define KC    32      // K chunk staged through LDS in kernel 1
#define LDSP  36      // padded LDS row stride (floats): bank = (4*row + k) % 64, conflict-free
#define SCALING 0.5f  // 8.0 / 16

// ---------------------------------------------------------------------------
// Kernel 0: build Wbig[144][4096] = concat(A as (128,4096), router_w (8,4096), zeros(8,4096))
// ---------------------------------------------------------------------------
__global__ void k_prep(const float* __restrict__ A, const float* __restrict__ rw,
                       float* __restrict__ Wbig) {
    int f4 = blockIdx.x * blockDim.x + threadIdx.x;   // float4 index, grid covers exactly 144*4096/4
    const int totalA  = (NE * NR * DIM) / 4;          // 131072
    const int totalRW = (NE * DIM) / 4;               // 8192
    float4 v = make_float4(0.f, 0.f, 0.f, 0.f);
    if (f4 < totalA)                 v = ((const float4*)A)[f4];
    else if (f4 < totalA + totalRW)  v = ((const float4*)rw)[f4 - totalA];
    ((float4*)Wbig)[f4] = v;
}

// ---------------------------------------------------------------------------
// Kernel 1: Y[N,144] = x @ Wbig^T   (fused h + router logits; x read exactly once)
// block = 256 threads = 8 waves; each wave owns a 16-row stripe, all 9 col tiles.
// ---------------------------------------------------------------------------
__global__ __launch_bounds__(256) void k_hlogits(const float* __restrict__ x,
                                                 const float* __restrict__ Wbig,
                                                 float* __restrict__ Y) {
    __shared__ float xs[128 * LDSP];   // 18.4 KB
    __shared__ float wsh[CW * LDSP];   // 20.7 KB
    const int t    = threadIdx.x;
    const int lane = t & 31;
    const int wid  = t >> 5;           // wave 0..7
    const int half = lane >> 4;        // 0 or 1
    const int ln   = lane & 15;
    const int rowBase = blockIdx.x * 128;

    const v8f vzero = {0.f,0.f,0.f,0.f,0.f,0.f,0.f,0.f};
    v8f acc[9];
#pragma unroll
    for (int i = 0; i < 9; ++i) acc[i] = vzero;

    for (int k0 = 0; k0 < DIM; k0 += KC) {
        __syncthreads();   // protect LDS from previous iteration's consumers
        // stage x block: 128 rows x 32 floats = 1024 float4, coalesced
        for (int f = t; f < 1024; f += 256) {
            int row = f >> 3, seg = f & 7;
            float4 v = *(const float4*)(x + (size_t)(rowBase + row) * DIM + k0 + seg * 4);
            *(float4*)(xs + row * LDSP + seg * 4) = v;
        }
        // stage W chunk: 144 rows x 32 floats = 1152 float4
        for (int f = t; f < 1152; f += 256) {
            int row = f >> 3, seg = f & 7;
            float4 v = *(const float4*)(Wbig + (size_t)row * DIM + k0 + seg * 4);
            *(float4*)(wsh + row * LDSP + seg * 4) = v;
        }
        __syncthreads();
#pragma unroll
        for (int ks = 0; ks < KC / 4; ++ks) {
            const int kk = ks * 4 + half * 2;
            // A frag 16x4 f32: vgpr0 = x[M=ln, K=half*2], vgpr1 = K+1
            v2f a = *(const v2f*)(xs + (wid * 16 + ln) * LDSP + kk);
#pragma unroll
            for (int tile = 0; tile < 9; ++tile) {
                // B frag 4x16: vgpr0 = BigB[K=half*2, N=ln] = Wbig[col, k]
                v2f b = *(const v2f*)(wsh + (tile * 16 + ln) * LDSP + kk);
                acc[tile] = __builtin_amdgcn_wmma_f32_16x16x4_f32(
                    false, a, false, b, (short)0, acc[tile], false, false);
            }
        }
    }
    // store: C/D layout: VGPR i -> M = i + half*8, N = ln
    const int row0 = rowBase + wid * 16;
#pragma unroll
    for (int tile = 0; tile < 9; ++tile) {
        int col = tile * 16 + ln;
        if (col < 136) {
#pragma unroll
            for (int i = 0; i < 8; ++i)
                Y[(size_t)(row0 + i + half * 8) * CW + col] = acc[tile][i];
        }
    }
}

// ---------------------------------------------------------------------------
// Kernel 2: gates. softmax over 8 logits, top-2 renormalized -> dense w[N,8]
// ---------------------------------------------------------------------------
__global__ void k_gate(const float* __restrict__ Y, const float* __restrict__ rb,
                       float* __restrict__ w) {
    int n = blockIdx.x * blockDim.x + threadIdx.x;
    if (n >= NTOK) return;
    float lg[NE];
    float m = -1e30f;
#pragma unroll
    for (int e = 0; e < NE; ++e) {
        lg[e] = Y[(size_t)n * CW + 128 + e] + rb[e];
        m = fmaxf(m, lg[e]);
    }
    float p[NE];
#pragma unroll
    for (int e = 0; e < NE; ++e) p[e] = __expf(lg[e] - m);
    // top-2 (softmax is monotonic, so argmax over p == argmax over probs)
    int i1 = 0;
#pragma unroll
    for (int e = 1; e < NE; ++e) if (p[e] > p[i1]) i1 = e;
    int i2 = (i1 == 0) ? 1 : 0;
#pragma unroll
    for (int e = 0; e < NE; ++e) if (e != i1 && p[e] > p[i2]) i2 = e;
    float s2 = p[i1] + p[i2];
    float g1 = p[i1] / s2, g2 = p[i2] / s2;
#pragma unroll
    for (int e = 0; e < NE; ++e) {
        float v = (e == i1) ? g1 : ((e == i2) ? g2 : 0.f);
        w[(size_t)n * NE + e] = v;
    }
}

// ---------------------------------------------------------------------------
// Kernel 3: out[N,4096] = (w ⊙ h * SCALING) @ BigB, BigB[e*16+r, o] = Bm[e,o,r]
// wave hoists 32 gate-scaled A-frags (whole K=128) once, sweeps 64 col tiles.
// ---------------------------------------------------------------------------
__global__ __launch_bounds__(256) void k_out(const float* __restrict__ Y,
                                             const float* __restrict__ w,
                                             const float* __restrict__ Bm,
                                             float* __restrict__ out) {
    const int t    = threadIdx.x;
    const int lane = t & 31;
    const int wid  = t >> 5;
    const int half = lane >> 4;
    const int ln   = lane & 15;
    const int row0 = blockIdx.x * 128 + wid * 16;
    const int row  = row0 + ln;        // this lane's token row for A frags

    float gv[NE];
#pragma unroll
    for (int e = 0; e < NE; ++e) gv[e] = w[(size_t)row * NE + e] * SCALING;

    v2f a[32];
#pragma unroll
    for (int ks = 0; ks < 32; ++ks) {
        const int k = ks * 4 + half * 2;
        v2f h2 = *(const v2f*)(Y + (size_t)row * CW + k);
        const float g = gv[ks >> 2];   // e = k >> 4 = ks >> 2
        a[ks] = h2 * g;
    }

    const int ct0 = blockIdx.y * 64;
    const v8f vzero = {0.f,0.f,0.f,0.f,0.f,0.f,0.f,0.f};
    for (int ct = ct0; ct < ct0 + 64; ++ct) {
        const int col = ct * 16 + ln;
        v8f acc = vzero;
#pragma unroll
        for (int ks = 0; ks < 32; ++ks) {
            const int e = ks >> 2;
            const int r = (ks & 3) * 4 + half * 2;      // k & 15
            v2f b = *(const v2f*)(Bm + ((size_t)e * NO + col) * NR + r);
            acc = __builtin_amdgcn_wmma_f32_16x16x4_f32(
                false, a[ks], false, b, (short)0, acc, false, false);
        }
#pragma unroll
        for (int i = 0; i < 8; ++i)
            out[(size_t)(row0 + i + half * 8) * NO + col] = acc[i];
    }
}

// ---------------------------------------------------------------------------
extern "C" void kernel_launch(void* const* d_in, const int* in_sizes, int n_in,
                              void* d_out, int out_size, void* d_ws, size_t ws_size,
                              hipStream_t stream) {
    (void)in_sizes; (void)n_in; (void)out_size; (void)ws_size;
    const float* x  = (const float*)d_in[0];
    const float* rw = (const float*)d_in[1];
    const float* rb = (const float*)d_in[2];
    const float* A  = (const float*)d_in[3];
    const float* Bm = (const float*)d_in[4];
    float* out = (float*)d_out;

    float* Wbig = (float*)d_ws;                         // 144*4096            = 0.59M floats
    float* Y    = Wbig + (size_t)CW * DIM;              // 16384*144           = 2.36M floats
    float* wg   = Y + (size_t)NTOK * CW;                // 16384*8             = 0.13M floats

    k_prep   <<<576, 256, 0, stream>>>(A, rw, Wbig);
    k_hlogits<<<128, 256, 0, stream>>>(x, Wbig, Y);
    k_gate   <<<64,  256, 0, stream>>>(Y, rb, wg);
    k_out    <<<dim3(128, 4), 256, 0, stream>>>(Y, wg, Bm, out);
}